// ResidualBlock_73014444032656
// MI455X (gfx1250) — compile-verified
//
#include <hip/hip_runtime.h>
#include <math.h>

// ---------------------------------------------------------------------------
// GMMConv residual block for MI455X (gfx1250, wave32).
// Dense projections via V_WMMA_F32_16X16X4_F32 (M templated so all B/C
// addresses fold into instruction immediates; B fragments double-buffered so
// the 16-WMMA chain per tile runs without per-op load waits).
// Edge phase: L2-resident gathers + global_atomic_add_f32 scatters.
// ---------------------------------------------------------------------------

typedef float v2f __attribute__((ext_vector_type(2)));
typedef float v8f __attribute__((ext_vector_type(8)));

#define CH 64          // Cin == Cout == 64
#define EPSG 1e-15f    // GMM sigma epsilon
#define BN_EPS 1e-5f

static inline int cdiv(int a, int b) { return (a + b - 1) / b; }

// ---------------------------------------------------------------------------
// C[N,M] = A[N,64] @ B[64,M]   (row-major, M,N multiples of 16; M constexpr)
// One wave per 16-row tile. A fragments (full K=64) cached in 32 VGPRs and
// reused across all column tiles; B fragments software-pipelined (tile nt+1
// fetched during tile nt's 16 chained v_wmma_f32_16x16x4_f32 ops).
//
// Fragment layouts (ISA 7.12.2):
//   A 16x4 : lane L -> row L&15, half=(L>>4): vgpr0 = K(half*2+0), vgpr1 = K(half*2+1)
//   B 4x16 : lane L -> col L&15, vgpr v -> K row (v + half*2)
//   C 16x16: lane L -> col L&15, vgpr v -> row (v + half*8)
// ---------------------------------------------------------------------------
template <int M>
__global__ __launch_bounds__(128) void gemm64_wmma(
    const float* __restrict__ A, const float* __restrict__ B,
    float* __restrict__ C, int rowTiles) {
  const int wave = threadIdx.x >> 5;
  const int tile = blockIdx.x * (blockDim.x >> 5) + wave;
  if (tile >= rowTiles) return;
  const int lane = threadIdx.x & 31;
  const int half = lane >> 4;
  const int mrow = lane & 15;
  const int r0 = tile * 16;

  // --- A fragments: full K=64 for this wave's 16 rows (32 VGPRs) ---
  const float* arow = A + (size_t)(r0 + mrow) * CH + half * 2;
  v2f a[16];
#pragma unroll
  for (int kk = 0; kk < 16; ++kk) {
    a[kk].x = arow[kk * 4 + 0];
    a[kk].y = arow[kk * 4 + 1];
  }

  // Lane-base pointers: all remaining offsets are compile-time immediates.
  const float* bptr = B + (size_t)(half * 2) * M + mrow;
  float* cptr = C + (size_t)(r0 + half * 8) * M + mrow;

  constexpr int NT = M / 16;
  v2f b[16];
#pragma unroll
  for (int kk = 0; kk < 16; ++kk) {           // prologue: tile 0 B fragments
    b[kk].x = bptr[kk * 4 * M];
    b[kk].y = bptr[kk * 4 * M + M];
  }

#pragma unroll
  for (int nt = 0; nt < NT; ++nt) {
    v2f bn[16];
    if (nt + 1 < NT) {                        // prefetch next tile's B frags
#pragma unroll
      for (int kk = 0; kk < 16; ++kk) {
        bn[kk].x = bptr[kk * 4 * M + (nt + 1) * 16];
        bn[kk].y = bptr[kk * 4 * M + M + (nt + 1) * 16];
      }
    }
    v8f c = {};
#pragma unroll
    for (int kk = 0; kk < 16; ++kk)
      c = __builtin_amdgcn_wmma_f32_16x16x4_f32(
          /*neg_a=*/false, a[kk], /*neg_b=*/false, b[kk],
          /*c_mod=*/(short)0, c, /*reuse_a=*/false, /*reuse_b=*/false);
#pragma unroll
    for (int v = 0; v < 8; ++v)
      cptr[(size_t)v * M + nt * 16] = c[v];
#pragma unroll
    for (int kk = 0; kk < 16; ++kk) b[kk] = bn[kk];
  }
}

// ---------------------------------------------------------------------------
// Edge phase: one wave per edge. Gaussians (K<=5, dim=1) computed per lane,
// gather float2 per (k,lane) from xp[src] (L2-resident), weighted sum,
// atomic scatter to dst.
// ---------------------------------------------------------------------------
template <int K>
__global__ __launch_bounds__(256) void edge_agg(
    const int* __restrict__ ei, int E, const float* __restrict__ eattr,
    const float* __restrict__ mu, const float* __restrict__ sig,
    const float* __restrict__ xp, float* __restrict__ agg) {
  const int wave = threadIdx.x >> 5;
  const int e = blockIdx.x * (blockDim.x >> 5) + wave;
  if (e >= E) return;
  const int lane = threadIdx.x & 31;

  const int src = ei[e];
  const int dst = ei[E + e];
  const float p = eattr[e];

  float g[K];
#pragma unroll
  for (int k = 0; k < K; ++k) {
    const float d = p - mu[k];
    const float s = sig[k];
    g[k] = expf(-0.5f * d * d / (EPSG + s * s));
  }

  const float2* row = (const float2*)(xp + (size_t)src * (K * CH));
  float mx = 0.f, my = 0.f;
#pragma unroll
  for (int k = 0; k < K; ++k) {
    const float2 v = row[k * (CH / 2) + lane];
    mx += g[k] * v.x;
    my += g[k] * v.y;
  }
  float* out = agg + (size_t)dst * CH + 2 * lane;
  atomicAdd(out + 0, mx);
  atomicAdd(out + 1, my);
}

__global__ void deg_count(const int* __restrict__ ei, int E,
                          float* __restrict__ deg) {
  const int e = blockIdx.x * blockDim.x + threadIdx.x;
  if (e < E) atomicAdd(deg + ei[E + e], 1.0f);
}

__global__ void zero_f32(float* __restrict__ p, size_t n) {
  for (size_t i = blockIdx.x * (size_t)blockDim.x + threadIdx.x; i < n;
       i += (size_t)gridDim.x * blockDim.x)
    p[i] = 0.0f;
}

// pre = agg/clamp(deg,1) + xroot + bias
__global__ void finalize_pre(const float* __restrict__ agg,
                             const float* __restrict__ deg,
                             const float* __restrict__ xr,
                             const float* __restrict__ bias,
                             float* __restrict__ out, int n) {
  const int i = blockIdx.x * blockDim.x + threadIdx.x;
  if (i >= n) return;
  const int node = i >> 6, c = i & 63;
  out[i] = agg[i] / fmaxf(deg[node], 1.0f) + xr[i] + bias[c];
}

// channel sums / sums-of-squares; grid stride is a multiple of 64 so each
// thread's channel is fixed -> local accumulate, one atomicAdd pair at end.
__global__ __launch_bounds__(256) void bn_sums(const float* __restrict__ x,
                                               int total,
                                               float* __restrict__ stats) {
  const int tid = blockIdx.x * blockDim.x + threadIdx.x;
  const int c = tid & 63;
  float s = 0.f, q = 0.f;
  for (int i = tid; i < total; i += gridDim.x * blockDim.x) {
    const float v = x[i];
    s += v;
    q += v * v;
  }
  atomicAdd(stats + c, s);
  atomicAdd(stats + 64 + c, q);
}

// coef[c] = gamma*rsqrt(var+eps); coef[64+c] = beta - mean*coef[c]
__global__ void bn_coef(const float* __restrict__ stats,
                        const float* __restrict__ gamma,
                        const float* __restrict__ beta, float invN,
                        float* __restrict__ coef) {
  const int c = threadIdx.x;
  if (c >= 64) return;
  const float m = stats[c] * invN;
  const float var = stats[64 + c] * invN - m * m;
  const float sc = gamma[c] * rsqrtf(var + BN_EPS);
  coef[c] = sc;
  coef[64 + c] = beta[c] - m * sc;
}

__global__ void apply_bn_elu(float* __restrict__ x,
                             const float* __restrict__ coef, int n) {
  const int i = blockIdx.x * blockDim.x + threadIdx.x;
  if (i >= n) return;
  const int c = i & 63;
  const float v = x[i] * coef[c] + coef[64 + c];
  x[i] = v > 0.f ? v : (expf(v) - 1.f);
}

// out = elu(bn2(h2pre) + bns(scpre))
__global__ void final_out(const float* __restrict__ h2,
                          const float* __restrict__ sc,
                          const float* __restrict__ coef2,
                          const float* __restrict__ coefs,
                          float* __restrict__ out, int n) {
  const int i = blockIdx.x * blockDim.x + threadIdx.x;
  if (i >= n) return;
  const int c = i & 63;
  const float v = h2[i] * coef2[c] + coef2[64 + c] +
                  sc[i] * coefs[c] + coefs[64 + c];
  out[i] = v > 0.f ? v : (expf(v) - 1.f);
}

// ---------------------------------------------------------------------------
extern "C" void kernel_launch(void* const* d_in, const int* in_sizes, int n_in,
                              void* d_out, int out_size, void* d_ws,
                              size_t ws_size, hipStream_t stream) {
  const float* x     = (const float*)d_in[0];
  const int*   ei    = (const int*)d_in[1];   // [2,E] (JAX default x64-off -> int32)
  const float* eattr = (const float*)d_in[2];
  const float* g1    = (const float*)d_in[3];
  const float* mu1   = (const float*)d_in[4];
  const float* sig1  = (const float*)d_in[5];
  const float* root1 = (const float*)d_in[6];
  const float* b1    = (const float*)d_in[7];
  const float* gam1  = (const float*)d_in[8];
  const float* bet1  = (const float*)d_in[9];
  const float* g2    = (const float*)d_in[10];
  const float* mu2   = (const float*)d_in[11];
  const float* sig2  = (const float*)d_in[12];
  const float* root2 = (const float*)d_in[13];
  const float* b2    = (const float*)d_in[14];
  const float* gam2  = (const float*)d_in[15];
  const float* bet2  = (const float*)d_in[16];
  const float* gs    = (const float*)d_in[17];
  const float* mus   = (const float*)d_in[18];
  const float* sigs  = (const float*)d_in[19];
  const float* roots = (const float*)d_in[20];
  const float* bs    = (const float*)d_in[21];
  const float* gams  = (const float*)d_in[22];
  const float* bets  = (const float*)d_in[23];
  float* out = (float*)d_out;

  const int N = in_sizes[0] / CH;  // 50000
  const int E = in_sizes[2];       // 800000 (edge_attr is [E,1])
  const int NC = N * CH;
  const int rowTiles = N / 16;     // 50000/16 = 3125

  // ---- workspace layout (floats) ----
  float* W = (float*)d_ws;
  size_t o = 0;
  float* xp    = W + o; o += (size_t)N * (5 * CH);  // projections (layer1 then layer2)
  float* xr    = W + o; o += (size_t)NC;            // root projection (reused)
  float* xps   = W + o; o += (size_t)NC;            // shortcut K=1 projection
  float* xrs   = W + o; o += (size_t)NC;            // shortcut root projection
  float* agg   = W + o; o += (size_t)NC;            // scatter accumulator (reused 3x)
  float* h     = W + o; o += (size_t)NC;            // layer1 activation
  float* h2pre = W + o; o += (size_t)NC;            // layer2 pre-BN
  float* scpre = W + o; o += (size_t)NC;            // shortcut pre-BN
  float* deg   = W + o; o += (size_t)N;
  float* stats = W + o; o += 128;                   // sum / sumsq scratch
  float* coef1 = W + o; o += 128;
  float* coef2 = W + o; o += 128;
  float* coefs = W + o; o += 128;
  (void)ws_size; (void)n_in; (void)out_size;

  const int gemmBlk = 128, gemmWaves = gemmBlk / 32;
  const int gemmGrid = cdiv(rowTiles, gemmWaves);
  const int edgeBlk = 256, edgeWaves = edgeBlk / 32;
  const int edgeGrid = cdiv(E, edgeWaves);
  const int ewGrid = cdiv(NC, 256);
  const float invN = 1.0f / (float)N;

  // ---- stage 0: node-level projections of x ----
  gemm64_wmma<5 * CH><<<gemmGrid, gemmBlk, 0, stream>>>(x, g1, xp, rowTiles);
  gemm64_wmma<CH><<<gemmGrid, gemmBlk, 0, stream>>>(x, root1, xr, rowTiles);
  gemm64_wmma<CH><<<gemmGrid, gemmBlk, 0, stream>>>(x, gs, xps, rowTiles);
  gemm64_wmma<CH><<<gemmGrid, gemmBlk, 0, stream>>>(x, roots, xrs, rowTiles);

  // ---- degrees (shared by all three convs) ----
  zero_f32<<<1024, 256, 0, stream>>>(deg, (size_t)N);
  deg_count<<<cdiv(E, 256), 256, 0, stream>>>(ei, E, deg);

  // ---- layer 1 ----
  zero_f32<<<2048, 256, 0, stream>>>(agg, (size_t)NC);
  edge_agg<5><<<edgeGrid, edgeBlk, 0, stream>>>(ei, E, eattr, mu1, sig1, xp, agg);
  finalize_pre<<<ewGrid, 256, 0, stream>>>(agg, deg, xr, b1, h, NC);
  zero_f32<<<1, 128, 0, stream>>>(stats, 128);
  bn_sums<<<512, 256, 0, stream>>>(h, NC, stats);
  bn_coef<<<1, 64, 0, stream>>>(stats, gam1, bet1, invN, coef1);
  apply_bn_elu<<<ewGrid, 256, 0, stream>>>(h, coef1, NC);

  // ---- layer 2 (input h) ----
  gemm64_wmma<5 * CH><<<gemmGrid, gemmBlk, 0, stream>>>(h, g2, xp, rowTiles);
  gemm64_wmma<CH><<<gemmGrid, gemmBlk, 0, stream>>>(h, root2, xr, rowTiles);
  zero_f32<<<2048, 256, 0, stream>>>(agg, (size_t)NC);
  edge_agg<5><<<edgeGrid, edgeBlk, 0, stream>>>(ei, E, eattr, mu2, sig2, xp, agg);
  finalize_pre<<<ewGrid, 256, 0, stream>>>(agg, deg, xr, b2, h2pre, NC);
  zero_f32<<<1, 128, 0, stream>>>(stats, 128);
  bn_sums<<<512, 256, 0, stream>>>(h2pre, NC, stats);
  bn_coef<<<1, 64, 0, stream>>>(stats, gam2, bet2, invN, coef2);

  // ---- shortcut (input x, K=1) ----
  zero_f32<<<2048, 256, 0, stream>>>(agg, (size_t)NC);
  edge_agg<1><<<edgeGrid, edgeBlk, 0, stream>>>(ei, E, eattr, mus, sigs, xps, agg);
  finalize_pre<<<ewGrid, 256, 0, stream>>>(agg, deg, xrs, bs, scpre, NC);
  zero_f32<<<1, 128, 0, stream>>>(stats, 128);
  bn_sums<<<512, 256, 0, stream>>>(scpre, NC, stats);
  bn_coef<<<1, 64, 0, stream>>>(stats, gams, bets, invN, coefs);

  // ---- residual + ELU ----
  final_out<<<ewGrid, 256, 0, stream>>>(h2pre, scpre, coef2, coefs, out, NC);
}